// RotaryPositionalEmbeddings_63788854280603
// MI455X (gfx1250) — compile-verified
//
#include <hip/hip_runtime.h>

// Rotary positional embedding matrix builder for MI455X (gfx1250).
// Output: R[2048, 256, 256] f32 = 512 MiB, ~99.4% zeros.
// Pure store-bandwidth problem: ideal ~= 512MiB / 23.3 TB/s ~= 23 us.
// Strategy: 1 thread per float4 group; contiguous 512B per wave32.
//  - band lanes (<= 2 per wave, even rows only): compute cos/sin, NT b128 store
//  - zero lanes: async store of a zero LDS slab via the CDNA5
//    global_store_async_from_lds_b128 path (ASYNCcnt; endpgm waits idle).

typedef float v4f __attribute__((ext_vector_type(4)));
typedef int   v4i __attribute__((ext_vector_type(4)));

// Pointer types matching the async-store builtin's signature:
// (v4i addrspace(1)*, v4i addrspace(3)*, Ii offset, Ii cpol)
typedef __attribute__((address_space(1))) v4i* glob_v4i_ptr;
typedef __attribute__((address_space(3))) v4i* lds_v4i_ptr;

#define BDIM   2048
#define EDIM   256
#define KITER  127          // D/2 - 1
#define NGROUP ((BDIM * EDIM * EDIM) / 4)   // 33,554,432 float4 groups

// log2(10000) / 128  (f32): theta_i = exp2(-(i-1) * L2T)
#define L2T 0.103810253f

__device__ __forceinline__ float theta_of(int i) {
    // theta_i = 10000^(-2*(i-1)/D) computed as exp2 — same lowering JAX uses for
    // f32 pow; within ~1 ulp, i.e. inside the reference's own angle noise floor.
    return exp2f(-(float)(i - 1) * L2T);
}

__global__ __launch_bounds__(256) void rope_matrix_fill(float* __restrict__ out) {
    // 512B of zeros in LDS, source for the async bulk-zero stores.
    __shared__ __align__(16) float zslab[128];
    if (threadIdx.x < 128) zslab[threadIdx.x] = 0.0f;
    __syncthreads();

    const unsigned gid = blockIdx.x * 256u + threadIdx.x;  // float4-group id
    const unsigned c4  = gid & 63u;                        // column group (4 cols)
    const unsigned r   = (gid >> 6) & 255u;                // row (wave-uniform)
    const unsigned p   = gid >> 14;                        // position

    v4f v = {0.0f, 0.0f, 0.0f, 0.0f};
    bool nz = false;

    if ((r & 1u) == 0u) {                 // even rows only; uniform per wave32
        const int   j    = (int)(r >> 1);                  // 0..127
        const int   base = (int)(c4 << 2);
        const float pf   = (float)p;
#pragma unroll
        for (int t = 0; t < 2; ++t) {                      // even cols base, base+2
            const int c = base + 2 * t;
            const int d = c - (int)r;
            float val = 0.0f;
            if (d == 0) {                                  // diagonal: cos
                const int jj = (j < KITER) ? j : (KITER - 1);
                val = cosf(pf * theta_of(jj));
            } else if (d == 2 && j <= KITER - 1) {         // upper band: -sin(theta_j)
                val = -sinf(pf * theta_of(j));
            } else if (d == -2 && j >= 1) {                // lower band: +sin(theta_{j-1})
                val = sinf(pf * theta_of(j - 1));
            }
            if (val != 0.0f) nz = true;
            v[2 * t] = val;
        }
    }

    float* dst = out + ((size_t)gid << 2);
    if (nz) {
        // band values: regular non-temporal 128-bit store
        __builtin_nontemporal_store(v, (v4f*)dst);
    } else {
#if defined(__gfx1250__) && __has_builtin(__builtin_amdgcn_global_store_async_from_lds_b128)
        // bulk zeros: CDNA5 async LDS->global store (b128 per lane, 512B/wave)
        __builtin_amdgcn_global_store_async_from_lds_b128(
            (glob_v4i_ptr)dst,
            (lds_v4i_ptr)&zslab[(threadIdx.x & 31u) * 4u],
            0, 0);
#else
        __builtin_nontemporal_store(v, (v4f*)dst);
#endif
    }
    // S_ENDPGM performs an implicit wait-idle (LOADcnt/STOREcnt/ASYNCcnt), so the
    // async stores are guaranteed complete before LDS deallocation / wave exit.
}

extern "C" void kernel_launch(void* const* d_in, const int* in_sizes, int n_in,
                              void* d_out, int out_size, void* d_ws, size_t ws_size,
                              hipStream_t stream) {
    // Inputs are the static scalars block_size=2048, embedding_dim=256 (device-
    // resident single-element arrays); the shape is compile-time in the reference,
    // so we bake it in. out_size == 2048*256*256.
    (void)d_in; (void)in_sizes; (void)n_in; (void)out_size; (void)d_ws; (void)ws_size;
    float* out = (float*)d_out;
    const unsigned nblocks = NGROUP / 256;   // 131072 blocks x 256 threads, 1 float4 each
    rope_matrix_fill<<<dim3(nblocks), dim3(256), 0, stream>>>(out);
}